// MPSClassifier_19344532701784
// MI455X (gfx1250) — compile-verified
//
#include <hip/hip_runtime.h>

typedef float v2f __attribute__((ext_vector_type(2)));
typedef float v8f __attribute__((ext_vector_type(8)));

#define MPS_SIZE   784
#define MPS_NLAB   10
#define MPS_BATCH  128

// ---------------------------------------------------------------------------
// State convention: the running 32x32 product P is kept in WMMA C/D layout:
//   reg tile (t,j), vgpr r, lane (ln,hi):  P[16*t + 8*hi + r][16*j + ln]
// Feeding these registers directly as the B operand presents row-permuted
// P[sigma(k)] to the MAC array, with sigma(4*kk+2*hi+d) = 2*kk+8*hi+d.
// We compensate exactly by loading the A matrix with sigma-permuted columns:
//   A position (chunk kk, half hi, elem d) <- M[row][16*k2 + 2*kk + 8*hi + d]
// Then D = M x P exactly, and D is again in C/D layout. No cross-lane ops.
// ---------------------------------------------------------------------------

// 32x32 product via 32 x v_wmma_f32_16x16x4_f32; result into plain float state.
__device__ __forceinline__ void mm32f(const float m[2][2][8], const float p[2][2][8],
                                      float out[2][2][8]) {
#pragma unroll
    for (int i = 0; i < 2; ++i) {
#pragma unroll
        for (int j = 0; j < 2; ++j) {
            v8f c = {};
#pragma unroll
            for (int k2 = 0; k2 < 2; ++k2) {
#pragma unroll
                for (int kk = 0; kk < 4; ++kk) {
                    v2f a;
                    a[0] = m[i][k2][2 * kk];
                    a[1] = m[i][k2][2 * kk + 1];
                    v2f bb;
                    bb[0] = p[k2][j][2 * kk];
                    bb[1] = p[k2][j][2 * kk + 1];
                    c = __builtin_amdgcn_wmma_f32_16x16x4_f32(
                        false, a, false, bb, (short)0, c, false, false);
                }
            }
#pragma unroll
            for (int r = 0; r < 8; ++r) out[i][j][r] = c[r];
        }
    }
}

// Identity in C-layout state convention.
__device__ __forceinline__ void identity_c(float p[2][2][8], int ln, int hi) {
#pragma unroll
    for (int t = 0; t < 2; ++t)
#pragma unroll
        for (int j = 0; j < 2; ++j)
#pragma unroll
            for (int r = 0; r < 8; ++r)
                p[t][j][r] = (t == j && (8 * hi + r) == ln) ? 1.0f : 0.0f;
}

// Site matrix M = core[s,:,0,:]*e0 + core[s,:,1,:]*e1, sigma-permuted A-layout.
__device__ __forceinline__ void load_site_perm(const float* __restrict__ core, int s,
                                               float e0, float e1, int ln, int hi,
                                               float m[2][2][8]) {
#pragma unroll
    for (int i = 0; i < 2; ++i) {
        const float* rb = core + ((size_t)s * 32 + (16 * i + ln)) * 64;
#pragma unroll
        for (int k2 = 0; k2 < 2; ++k2) {
#pragma unroll
            for (int kk = 0; kk < 4; ++kk) {
                int col = 16 * k2 + 2 * kk + 8 * hi;   // sigma-permuted column pair
                v2f c0 = *(const v2f*)(rb + col);       // emb d = 0
                v2f c1 = *(const v2f*)(rb + 32 + col);  // emb d = 1
                m[i][k2][2 * kk]     = c0[0] * e0 + c1[0] * e1;
                m[i][k2][2 * kk + 1] = c0[1] * e0 + c1[1] * e1;
            }
        }
    }
}

// Row-major 32x32 -> sigma-permuted A-layout.
__device__ __forceinline__ void load_a_perm(const float* __restrict__ src, int ln, int hi,
                                            float m[2][2][8]) {
#pragma unroll
    for (int i = 0; i < 2; ++i) {
        const float* rb = src + (size_t)(16 * i + ln) * 32;
#pragma unroll
        for (int k2 = 0; k2 < 2; ++k2) {
#pragma unroll
            for (int kk = 0; kk < 4; ++kk) {
                v2f v = *(const v2f*)(rb + 16 * k2 + 2 * kk + 8 * hi);
                m[i][k2][2 * kk]     = v[0];
                m[i][k2][2 * kk + 1] = v[1];
            }
        }
    }
}

// Row-major 32x32 -> C-layout state.
__device__ __forceinline__ void load_c_rm(const float* __restrict__ src, int ln, int hi,
                                          float p[2][2][8]) {
#pragma unroll
    for (int t = 0; t < 2; ++t)
#pragma unroll
        for (int j = 0; j < 2; ++j)
#pragma unroll
            for (int r = 0; r < 8; ++r)
                p[t][j][r] = src[(size_t)(16 * t + 8 * hi + r) * 32 + 16 * j + ln];
}

// C-layout state -> row-major 32x32.
__device__ __forceinline__ void store_c_rm(float* __restrict__ dst, int ln, int hi,
                                           const float p[2][2][8]) {
#pragma unroll
    for (int t = 0; t < 2; ++t)
#pragma unroll
        for (int j = 0; j < 2; ++j)
#pragma unroll
            for (int r = 0; r < 8; ++r)
                dst[(size_t)(16 * t + 8 * hi + r) * 32 + 16 * j + ln] = p[t][j][r];
}

// One chain step: pout = M_s * pin.
__device__ __forceinline__ void site_step(const float* __restrict__ xrow,
                                          const float* __restrict__ core, int s,
                                          int ln, int hi,
                                          const float pin[2][2][8], float pout[2][2][8]) {
    float x = xrow[s];                               // scalar (uniform) load
    float m[2][2][8];
    load_site_perm(core, s, 1.0f - x, x, ln, hi, m);
    mm32f(m, pin, pout);
}

// ---------------- Kernel 1: per-(batch, segment) product of seg_len site matrices
__global__ void mps_seg_kernel(const float* __restrict__ xin,
                               const float* __restrict__ core,
                               float* __restrict__ ws, int nseg, int seg_len) {
    int wid = __builtin_amdgcn_readfirstlane(
        (blockIdx.x * blockDim.x + threadIdx.x) >> 5);
    int lane = threadIdx.x & 31;
    int ln = lane & 15, hi = lane >> 4;
    int b = wid / nseg;
    int seg = wid - b * nseg;
    const float* xrow = xin + (size_t)b * MPS_SIZE;

    int s = seg * seg_len + seg_len - 1;
    int npairs = seg_len >> 1;

    float p[2][2][8], q[2][2][8];
    if (seg_len & 1) {            // uniform branch: EXEC stays full
        identity_c(p, ln, hi);
        site_step(xrow, core, s, ln, hi, p, q);
        --s;
    } else {
        identity_c(q, ln, hi);
    }
    for (int it = 0; it < npairs; ++it) {   // ping-pong: no state copies
        site_step(xrow, core, s,     ln, hi, q, p);
        site_step(xrow, core, s - 1, ln, hi, p, q);
        s -= 2;
    }
    store_c_rm(ws + ((size_t)wid << 10), ln, hi, q);
}

// ---------------- Kernel 2: combine nseg/2 segment products -> left / right ---
__global__ void mps_combine_kernel(float* __restrict__ ws, int nseg) {
    int wid = __builtin_amdgcn_readfirstlane(
        (blockIdx.x * blockDim.x + threadIdx.x) >> 5);
    int lane = threadIdx.x & 31;
    int ln = lane & 15, hi = lane >> 4;
    int b = wid >> 1;
    int half = wid & 1;
    int half_n = nseg >> 1;                 // 8 or 14: always even
    float* base = ws + ((size_t)(b * nseg + half * half_n) << 10);

    float p[2][2][8], q[2][2][8], m[2][2][8];
    identity_c(q, ln, hi);
    for (int i = half_n - 1; i > 0; i -= 2) {
        load_a_perm(base + ((size_t)i << 10), ln, hi, m);
        mm32f(m, q, p);
        load_a_perm(base + ((size_t)(i - 1) << 10), ln, hi, m);
        mm32f(m, p, q);
    }
    store_c_rm(base, ln, hi, q);  // left -> slot b*nseg, right -> slot b*nseg+half_n
}

// ---------------- Kernel 3: G = right*left, out[l] = sum T_l[j,k] * G[k,j] ----
__global__ void mps_trace_kernel(const float* __restrict__ ws,
                                 const float* __restrict__ lab,
                                 float* __restrict__ out, int nseg) {
    int wid = __builtin_amdgcn_readfirstlane(
        (blockIdx.x * blockDim.x + threadIdx.x) >> 5);
    int lane = threadIdx.x & 31;
    int ln = lane & 15, hi = lane >> 4;
    int b = wid;

    float m[2][2][8], p[2][2][8], g[2][2][8];
    load_a_perm(ws + ((size_t)(b * nseg + (nseg >> 1)) << 10), ln, hi, m);  // right (A)
    load_c_rm(ws + ((size_t)(b * nseg) << 10), ln, hi, p);                  // left  (B)
    mm32f(m, p, g);  // G = right * left, C-layout

#pragma unroll
    for (int t = 0; t < MPS_NLAB; ++t) {
        float acc = 0.0f;
#pragma unroll
        for (int i = 0; i < 2; ++i)
#pragma unroll
            for (int j = 0; j < 2; ++j)
#pragma unroll
                for (int r = 0; r < 8; ++r) {
                    int k = 16 * i + 8 * hi + r;   // row of G
                    int c = 16 * j + ln;           // col of G
                    acc += g[i][j][r] * lab[((size_t)c * MPS_NLAB + t) * 32 + k];
                }
#pragma unroll
        for (int off = 16; off > 0; off >>= 1) acc += __shfl_xor(acc, off, 32);
        if (lane == 0) out[b * MPS_NLAB + t] = acc;
    }
}

extern "C" void kernel_launch(void* const* d_in, const int* in_sizes, int n_in,
                              void* d_out, int out_size, void* d_ws, size_t ws_size,
                              hipStream_t stream) {
    const float* xin  = (const float*)d_in[0];  // [128, 784]
    const float* core = (const float*)d_in[1];  // [784, 32, 2, 32]
    const float* lab  = (const float*)d_in[2];  // [32, 10, 32]
    float* out = (float*)d_out;                 // [128, 10]
    float* ws  = (float*)d_ws;

    // Prefer 28 segments of 28 sites (3584 waves, shorter serial chains) if the
    // workspace allows; otherwise 16 segments of 49 (8 MB).
    size_t need28 = (size_t)MPS_BATCH * 28 * 1024 * sizeof(float);
    int nseg = (ws_size >= need28) ? 28 : 16;
    int seg_len = (nseg == 28) ? 28 : 49;

    mps_seg_kernel<<<(MPS_BATCH * nseg) / 8, 256, 0, stream>>>(xin, core, ws, nseg, seg_len);
    mps_combine_kernel<<<(MPS_BATCH * 2) / 8, 256, 0, stream>>>(ws, nseg);
    mps_trace_kernel<<<MPS_BATCH / 4, 128, 0, stream>>>(ws, lab, out, nseg);
}